// STC_5231270167006
// MI455X (gfx1250) — compile-verified
//
#include <hip/hip_runtime.h>
#include <cmath>

#ifndef M_LN2
#define M_LN2 0.693147180559945309
#endif

namespace {
constexpr int T_N = 2000;
constexpr int B_N = 64;
constexpr int C_N = 256;
constexpr int L_N = 128;
constexpr int S_N = 2 * L_N + 1;      // 257 CTC lattice nodes
constexpr int NN  = S_N + L_N + 1;    // 386 total nodes (CTC + star)
constexpr int NTHR = 448;             // 14 wave32 waves
constexpr float NEGF = -1.0e30f;      // finite -inf stand-in (matches reference)
}

// Generic (flat) pointer to a __shared__ object: on AMDGPU the shared->flat
// addrspacecast is {shared_aperture_hi32, lds_offset32}, so the low 32 bits
// are exactly the LDS byte address the async-LDS instructions want in VDST.
__device__ __forceinline__ unsigned lds_addr32(const void* p) {
  return (unsigned)(unsigned long long)p;
}

// CDNA5 async global->LDS copy (tracked with ASYNCcnt), GVS addressing:
// mem_addr = SGPR64 + VGPR32 + 0 ; LDS dest byte address in VDST VGPR.
__device__ __forceinline__ void async_row_b128(unsigned lds, unsigned gofs,
                                               unsigned long long sbase) {
  asm volatile("global_load_async_to_lds_b128 %0, %1, %2"
               :: "v"(lds), "v"(gofs), "s"(sbase) : "memory");
}
__device__ __forceinline__ void wait_async1() { asm volatile("s_wait_asynccnt 1" ::: "memory"); }
__device__ __forceinline__ void wait_async0() { asm volatile("s_wait_asynccnt 0" ::: "memory"); }

__global__ __launch_bounds__(NTHR, 1)
void stc_forward_kernel(const float* __restrict__ inputs,   // [T,B,C] log-probs
                        const int*   __restrict__ targets,  // [B,L]
                        float*       __restrict__ losses,   // [B]
                        float logp) {                        // log token-insertion prob
  __shared__ __align__(16) float lp[3][C_N];   // triple-buffered emission rows
  __shared__ float alpha[2][NN];               // double-buffered forward scores
  __shared__ int   srcs[NN][3];                // incoming arc sources (-1 = none)
  __shared__ int   meta[NN];                   // (kind<<16)|token
  __shared__ int   tgt[L_N];

  const int tid  = threadIdx.x;
  const int lane = tid & 31;
  const int b    = blockIdx.x;

  // ---- prefetch emission rows t=0,1 (async global->LDS) ----
  if (tid < 64) {
    const unsigned gofs = (unsigned)(tid * 16);
    #pragma unroll
    for (int t0 = 0; t0 < 2; ++t0) {
      const unsigned long long g =
          (unsigned long long)(inputs + ((size_t)t0 * B_N + b) * C_N);
      async_row_b128(lds_addr32(&lp[t0][tid * 4]), gofs, g);
    }
  }

  for (int l = tid; l < L_N; l += NTHR) tgt[l] = targets[b * L_N + l];
  __syncthreads();

  // ---- build per-node arc table: all arcs into a node share label+weight ----
  // kind: 0=blank(lp[0]) 1=token(lp[tok]) 2=star\tok(neglse)+logp 3=full star(lse)+logp
  for (int d = tid; d < NN; d += NTHR) {
    int s0, s1, s2, kd, tk;
    if (d < S_N) {
      if ((d & 1) == 0) {                    // even CTC node 2l: blank
        int l = d >> 1;
        s0 = d; s1 = (d > 0) ? d - 1 : -1; s2 = S_N + l; kd = 0; tk = 0;
      } else {                               // odd CTC node 2l+1: token tgt[l]
        int l = (d - 1) >> 1;
        s0 = d - 1; s1 = (d >= 3) ? d - 2 : -1; s2 = S_N + l; kd = 1; tk = tgt[l];
      }
    } else {                                 // star node c_l
      int l = d - S_N;
      s0 = (l >= 1) ? 2 * l - 1 : -1; s1 = 2 * l; s2 = d;
      kd = (l < L_N) ? 2 : 3; tk = (l < L_N) ? tgt[l] : 0;
    }
    srcs[d][0] = s0; srcs[d][1] = s1; srcs[d][2] = s2;
    meta[d] = (kd << 16) | tk;
    alpha[0][d] = (d == 0) ? 0.0f : NEGF;
  }

  if (tid < 64) wait_async1();   // row 0 has landed (row 1 may still fly)
  __syncthreads();

  int cur = 0;
  for (int t = 0; t < T_N; ++t) {
    const float* __restrict__ lpc = lp[t % 3];
    const float* __restrict__ ac  = alpha[cur];
    float* __restrict__ an        = alpha[cur ^ 1];

    // waves 8..13: wave-local logsumexp over classes 1..C-1 (star emission)
    float wlse = 0.0f;
    if (tid >= 256) {
      float v[8];
      float mx = NEGF;
      #pragma unroll
      for (int i = 0; i < 8; ++i) {
        const int c = lane + i * 32;
        float x = lpc[c];
        if (c == 0) x = NEGF;                 // blank excluded from star
        v[i] = x;
        mx = fmaxf(mx, x);
      }
      #pragma unroll
      for (int o = 16; o; o >>= 1) mx = fmaxf(mx, __shfl_xor(mx, o, 32));
      float s = 0.0f;
      #pragma unroll
      for (int i = 0; i < 8; ++i) s += __expf(v[i] - mx);
      #pragma unroll
      for (int o = 16; o; o >>= 1) s += __shfl_xor(s, o, 32);
      wlse = mx + __logf(s);
    }

    // one node per thread: alpha'[d] = LSE3(alpha[srcs]) + em[label_d](+logp)
    if (tid < NN) {
      const int md = meta[tid];
      const int kd = md >> 16;
      const int tk = md & 0xffff;
      float c;
      if (kd == 0)      c = lpc[0];
      else if (kd == 1) c = lpc[tk];
      else if (kd == 2) c = wlse + log1pf(1e-7f - __expf(lpc[tk] - wlse)) + logp;
      else              c = wlse + logp;
      const int s0 = srcs[tid][0], s1 = srcs[tid][1], s2 = srcs[tid][2];
      const float v0 = ((s0 >= 0) ? ac[s0] : NEGF) + c;
      const float v1 = ((s1 >= 0) ? ac[s1] : NEGF) + c;
      const float v2 = ac[s2] + c;
      float m = fmaxf(fmaxf(fmaxf(v0, v1), v2), NEGF);
      an[tid] = m + __logf(__expf(v0 - m) + __expf(v1 - m) + __expf(v2 - m) + 1e-38f);
    }

    // prefetch row t+2 into the buffer not referenced this step, then make
    // sure row t+1 has landed before anyone passes the barrier
    if (tid < 64) {
      if (t + 2 < T_N) {
        const unsigned long long g =
            (unsigned long long)(inputs + ((size_t)(t + 2) * B_N + b) * C_N);
        async_row_b128(lds_addr32(&lp[(t + 2) % 3][tid * 4]),
                       (unsigned)(tid * 16), g);
        wait_async1();                        // all rows <= t+1 complete
      } else {
        wait_async0();
      }
    }
    __syncthreads();                          // single barrier per time step
    cur ^= 1;
  }

  // loss_b = -logsumexp(alpha_T[{2L, 2L-1, S+L}])
  if (tid == 0) {
    const float* aT = alpha[cur];
    const float a0 = aT[2 * L_N];
    const float a1 = aT[2 * L_N - 1];
    const float a2 = aT[S_N + L_N];
    const float m  = fmaxf(fmaxf(a0, a1), a2);
    losses[b] = -(m + __logf(__expf(a0 - m) + __expf(a1 - m) + __expf(a2 - m)));
  }
}

// Deterministic ordered mean over the 64 per-batch losses.
__global__ void stc_reduce_kernel(const float* __restrict__ losses,
                                  float* __restrict__ out) {
  if (threadIdx.x == 0) {
    float s = 0.0f;
    for (int i = 0; i < B_N; ++i) s += losses[i];
    out[0] = s * (1.0f / (float)B_N);
  }
}

extern "C" void kernel_launch(void* const* d_in, const int* in_sizes, int n_in,
                              void* d_out, int out_size, void* d_ws, size_t ws_size,
                              hipStream_t stream) {
  (void)in_sizes; (void)n_in; (void)out_size; (void)ws_size;
  const float* inputs = (const float*)d_in[0];   // [T,B,C] float32 log-probs
  const int*   targets = (const int*)d_in[1];    // [B,L] int32
  float* losses = (float*)d_ws;                  // 64 floats of scratch

  // token-insertion penalty: prob = PLAST + (P0-PLAST)*exp(-NSTEP*ln2/THALF)
  const double prob = 0.1 + (1.0 - 0.1) * std::exp(-1.0 * M_LN2 / 100.0);
  const float logp = (float)std::log(prob);

  stc_forward_kernel<<<B_N, NTHR, 0, stream>>>(inputs, targets, losses, logp);
  stc_reduce_kernel<<<1, 32, 0, stream>>>(losses, (float*)d_out);
}